// MultiheadAttentionAttrExtract_28527172780170
// MI455X (gfx1250) — compile-verified
//
#include <hip/hip_runtime.h>
#include <math.h>

// ---------------- problem constants ----------------
constexpr int B_  = 8;
constexpr int T_  = 8;
constexpr int L_  = 197;
constexpr int E_  = 768;
constexpr int H_  = 12;
constexpr int QS_ = 16;
constexpr int C_  = 64;
constexpr int MX_ = B_ * T_ * L_;     // 12608 rows for x
constexpr float NEG_INF = -1.0e30f;

typedef float v2f __attribute__((ext_vector_type(2)));
typedef float v8f __attribute__((ext_vector_type(8)));

// D = A(16x4) * B(4x16) + C, fp32 WMMA (CDNA5 V_WMMA_F32_16X16X4_F32)
__device__ __forceinline__ v8f wmma4(v2f a, v2f b, v8f c) {
  return __builtin_amdgcn_wmma_f32_16x16x4_f32(false, a, false, b, (short)0, c,
                                               false, false);
}

// Async copy of 16 bytes global -> LDS (tracked by ASYNCcnt).
// ldsOff is the workgroup-relative LDS byte address (low 32 bits of a
// generic pointer to __shared__), gp a 16B-aligned global address.
__device__ __forceinline__ void async_copy_b128(uint32_t ldsOff,
                                                const float* gp) {
  asm volatile("global_load_async_to_lds_b128 %0, %1, off"
               :
               : "v"(ldsOff), "v"(gp)
               : "memory");
}
__device__ __forceinline__ void wait_async0() {
  asm volatile("s_wait_asynccnt 0x0" ::: "memory");
}
__device__ __forceinline__ uint32_t lds_addr(const void* p) {
  return (uint32_t)(uintptr_t)p;   // addr[31:0] of a shared ptr = LDS offset
}

// =====================================================================
// Generic GEMM:  out = A(M,K) @ W(N,K)^T + bias(N)
// Double-buffered async-to-LDS staging (GLOBAL_LOAD_ASYNC_TO_LDS_B128).
// Output columns may be split into up to 3 regions of width regionN
// (used to scatter qkv into the q/k/v slices of d_out).
// Block tile: 128(M) x 64(N), 8 waves as 4x2, each wave 32x32 (2x2 WMMA).
// LDS row stride 36 floats: 16B-aligned B128 stores, conflict-free frags.
// =====================================================================
__global__ __launch_bounds__(256) void gemm_bias_wmma(
    const float* __restrict__ A, const float* __restrict__ W,
    const float* __restrict__ bias, int M, int N, int K,
    float* __restrict__ out0, float* __restrict__ out1,
    float* __restrict__ out2, int regionN, int ldOut) {
  __shared__ float As[2][128][36];
  __shared__ float Ws[2][64][36];

  const int tid   = threadIdx.x;
  const int lane  = tid & 31;
  const int wave  = tid >> 5;
  const int warpM = wave >> 1;   // 0..3
  const int warpN = wave & 1;    // 0..1
  const int half  = lane >> 4;   // 0/1
  const int l15   = lane & 15;
  const int ko    = half * 2;    // A/B K-offset for this half

  const int blockM = blockIdx.y * 128;
  const int blockN = blockIdx.x * 64;

  v8f zero = {};
  v8f c[2][2];
  c[0][0] = zero; c[0][1] = zero; c[1][0] = zero; c[1][1] = zero;

  // issue async fills of one K-stage (32 wide) into buffer `buf`
  auto stage = [&](int k0, int buf) {
    // A tile 128x32: 4 x B128 per thread; clamp OOB rows (their C rows
    // are never stored, so duplicated data is harmless)
#pragma unroll
    for (int it = 0; it < 4; ++it) {
      int idx = tid + it * 256;          // 0..1023
      int r   = idx >> 3;                // 0..127
      int kq  = (idx & 7) << 2;          // 0,4,..,28
      int rg  = blockM + r;
      if (rg > M - 1) rg = M - 1;
      async_copy_b128(lds_addr(&As[buf][r][kq]),
                      A + (size_t)rg * K + k0 + kq);
    }
    // W tile 64x32: 2 x B128 per thread (N multiple of 64 -> no guard)
#pragma unroll
    for (int it = 0; it < 2; ++it) {
      int idx = tid + it * 256;          // 0..511
      int r   = idx >> 3;                // 0..63
      int kq  = (idx & 7) << 2;
      async_copy_b128(lds_addr(&Ws[buf][r][kq]),
                      W + (size_t)(blockN + r) * K + k0 + kq);
    }
  };

  const int nstage = K >> 5;             // K/32 (K is a multiple of 32)
  stage(0, 0);
  wait_async0();
  __syncthreads();

  for (int s = 0; s < nstage; ++s) {
    const int buf = s & 1;
    if (s + 1 < nstage) stage((s + 1) << 5, buf ^ 1);  // overlap with compute

#pragma unroll
    for (int kk = 0; kk < 32; kk += 4) {
      v2f a0, a1, b0, b1;
      const float* ap0 = &As[buf][warpM * 32 + l15][kk + ko];
      const float* ap1 = &As[buf][warpM * 32 + 16 + l15][kk + ko];
      const float* bp0 = &Ws[buf][warpN * 32 + l15][kk + ko];
      const float* bp1 = &Ws[buf][warpN * 32 + 16 + l15][kk + ko];
      a0.x = ap0[0]; a0.y = ap0[1];
      a1.x = ap1[0]; a1.y = ap1[1];
      b0.x = bp0[0]; b0.y = bp0[1];
      b1.x = bp1[0]; b1.y = bp1[1];
      c[0][0] = wmma4(a0, b0, c[0][0]);
      c[0][1] = wmma4(a0, b1, c[0][1]);
      c[1][0] = wmma4(a1, b0, c[1][0]);
      c[1][1] = wmma4(a1, b1, c[1][1]);
    }

    wait_async0();       // next-stage tiles fully in LDS
    __syncthreads();     // all waves done reading current buffers
  }

  // epilogue: bias + region scatter (region uniform per block: 64 | regionN)
  const int region  = blockN / regionN;
  const int colBase = blockN % regionN;
  float* outp = (region == 0) ? out0 : ((region == 1) ? out1 : out2);
#pragma unroll
  for (int tm = 0; tm < 2; ++tm) {
#pragma unroll
    for (int tn = 0; tn < 2; ++tn) {
      int nb  = blockN + warpN * 32 + tn * 16 + l15;   // global col for bias
      int col = colBase + warpN * 32 + tn * 16 + l15;  // col within region
      float bv = bias[nb];
#pragma unroll
      for (int i = 0; i < 8; ++i) {
        int m = blockM + warpM * 32 + tm * 16 + half * 8 + i;
        if (m < M) outp[(size_t)m * ldOut + col] = c[tm][tn][i] + bv;
      }
    }
  }
}

// =====================================================================
// Self-attention for x: one workgroup per (b*T+t, h).
// scores = softmax( (q*scale) @ k^T ) over keys; mix = P @ v.
// P converted C-layout -> A-layout through a small per-wave LDS chunk.
// =====================================================================
__global__ __launch_bounds__(256) void attn_kernel(
    const float* __restrict__ q, const float* __restrict__ k,
    const float* __restrict__ v, float* __restrict__ mix) {
  __shared__ float Pc[8][16 * 17];   // per-wave 16x16 P chunk (padded)

  const int bt   = blockIdx.x / H_;
  const int h    = blockIdx.x % H_;
  const int lane = threadIdx.x & 31;
  const int wave = threadIdx.x >> 5;
  const int half = lane >> 4;
  const int l15  = lane & 15;
  const int ko   = half * 2;
  const float scale = 0.125f;  // 1/sqrt(64)

  const float* qp = q + (size_t)bt * L_ * E_ + h * 64;
  const float* kp = k + (size_t)bt * L_ * E_ + h * 64;
  const float* vp = v + (size_t)bt * L_ * E_ + h * 64;
  float* Pl = &Pc[wave][0];
  v8f zero = {};

  for (int qt = wave; qt < 13; qt += 8) {   // 13 query tiles of 16 (208>=197)
    const int m0 = qt * 16;
    // A-operand: this lane owns q row (m0+l15); pre-scale by 1/sqrt(C)
    float aq[32];
    {
      int mrow = m0 + l15;
      bool mv  = mrow < L_;
      const float* qrow = qp + (size_t)mrow * E_;
#pragma unroll
      for (int i = 0; i < 16; ++i) {
        aq[2 * i]     = mv ? qrow[4 * i + ko] * scale : 0.f;
        aq[2 * i + 1] = mv ? qrow[4 * i + ko + 1] * scale : 0.f;
      }
    }

    // ---- scores: 13 key tiles, B-frags straight from global k ----
    v8f acc[13];
#pragma unroll
    for (int nt = 0; nt < 13; ++nt) acc[nt] = zero;
#pragma unroll
    for (int nt = 0; nt < 13; ++nt) {
      int ncol = nt * 16 + l15;
      if (ncol > L_ - 1) ncol = L_ - 1;           // clamp pad columns
      const float* krow = kp + (size_t)ncol * E_;
      if (nt + 1 < 13) {                          // prefetch next key row
        int nnext = (nt + 1) * 16 + l15;
        if (nnext > L_ - 1) nnext = L_ - 1;
        __builtin_prefetch(kp + (size_t)nnext * E_, 0, 3);
      }
#pragma unroll
      for (int kk = 0; kk < 16; ++kk) {
        v2f a, bf;
        a.x  = aq[2 * kk];
        a.y  = aq[2 * kk + 1];
        bf.x = krow[4 * kk + ko];
        bf.y = krow[4 * kk + ko + 1];
        acc[nt] = wmma4(a, bf, acc[nt]);
      }
    }
    // mask padded key columns (tile 12, cols 197..207)
    if (l15 >= 5) {
#pragma unroll
      for (int i = 0; i < 8; ++i) acc[12][i] = NEG_INF;
    }

    // ---- softmax across keys (row = i + 8*half, spread over 16 lanes) ----
#pragma unroll
    for (int i = 0; i < 8; ++i) {
      float mx = acc[0][i];
#pragma unroll
      for (int nt = 1; nt < 13; ++nt) mx = fmaxf(mx, acc[nt][i]);
#pragma unroll
      for (int off = 8; off > 0; off >>= 1) mx = fmaxf(mx, __shfl_xor(mx, off, 32));
      float s = 0.f;
#pragma unroll
      for (int nt = 0; nt < 13; ++nt) {
        float e = __expf(acc[nt][i] - mx);
        acc[nt][i] = e;
        s += e;
      }
#pragma unroll
      for (int off = 8; off > 0; off >>= 1) s += __shfl_xor(s, off, 32);
      float rinv = 1.f / s;
#pragma unroll
      for (int nt = 0; nt < 13; ++nt) acc[nt][i] *= rinv;
    }

    // ---- mix = P @ v : stream P chunks through LDS, v from global ----
    v8f mc[4];
#pragma unroll
    for (int ct = 0; ct < 4; ++ct) mc[ct] = zero;
#pragma unroll
    for (int jt = 0; jt < 13; ++jt) {
#pragma unroll
      for (int i = 0; i < 8; ++i)
        Pl[(half * 8 + i) * 17 + l15] = acc[jt][i];
      asm volatile("s_wait_dscnt 0" ::: "memory");
#pragma unroll
      for (int ct = 0; ct < 4; ++ct) {
        const float* vcol = vp + ct * 16 + l15;
#pragma unroll
        for (int kk = 0; kk < 4; ++kk) {
          int j0  = jt * 16 + kk * 4 + ko;
          int j0c = (j0 > L_ - 1) ? (L_ - 1) : j0;
          int j1c = (j0 + 1 > L_ - 1) ? (L_ - 1) : (j0 + 1);
          v2f a, bf;
          a.x  = Pl[l15 * 17 + kk * 4 + ko];
          a.y  = Pl[l15 * 17 + kk * 4 + ko + 1];
          bf.x = vcol[(size_t)j0c * E_];
          bf.y = vcol[(size_t)j1c * E_];
          mc[ct] = wmma4(a, bf, mc[ct]);
        }
      }
      asm volatile("s_wait_dscnt 0" ::: "memory");
    }

    // store mix rows (b,t,l,h,c) row-major as (MX_, E_)
#pragma unroll
    for (int ct = 0; ct < 4; ++ct) {
#pragma unroll
      for (int i = 0; i < 8; ++i) {
        int m = m0 + half * 8 + i;
        if (m < L_)
          mix[((size_t)bt * L_ + m) * E_ + h * 64 + ct * 16 + l15] = mc[ct][i];
      }
    }
  }
}

// =====================================================================
// s-path attention: one wave per (b*T+t, h). 16 queries x 196 keys.
// k' = k[:,:,1:]+te_k, v' = v[:,:,1:]+te_v, scores += patch_mask.
// Writes smix laid out as [(b*QS+q)*E + e] * T + t  (conv-ready).
// =====================================================================
__global__ __launch_bounds__(32) void s_attn_kernel(
    const float* __restrict__ sq, const float* __restrict__ k,
    const float* __restrict__ v, const float* __restrict__ te_zz,
    const float* __restrict__ pmask, float* __restrict__ smix) {
  __shared__ float Pl[16 * 17];

  const int bt   = blockIdx.x / H_;
  const int h    = blockIdx.x % H_;
  const int b    = bt / T_;
  const int t    = bt % T_;
  const int lane = threadIdx.x & 31;
  const int half = lane >> 4;
  const int l15  = lane & 15;
  const int ko   = half * 2;
  const float scale = 0.125f;
  const int NK = L_ - 1;  // 196 keys

  const float* sqp = sq + (size_t)b * QS_ * E_ + h * 64;
  const float* kp  = k + ((size_t)bt * L_ + 1) * E_ + h * 64;
  const float* vp  = v + ((size_t)bt * L_ + 1) * E_ + h * 64;
  const float* tek = te_zz + (size_t)t * 2304 + 768 + h * 64;
  const float* tev = te_zz + (size_t)t * 2304 + 1536 + h * 64;
  const float* pm  = pmask + (size_t)bt * NK;
  v8f zero = {};

  // A-operand: lane owns s_q row l15 (QS_=16 rows, all valid)
  float aq[32];
  {
    const float* qrow = sqp + (size_t)l15 * E_;
#pragma unroll
    for (int i = 0; i < 16; ++i) {
      aq[2 * i]     = qrow[4 * i + ko] * scale;
      aq[2 * i + 1] = qrow[4 * i + ko + 1] * scale;
    }
  }

  // ---- scores against k' = k + te_k ----
  v8f acc[13];
#pragma unroll
  for (int nt = 0; nt < 13; ++nt) acc[nt] = zero;
#pragma unroll
  for (int nt = 0; nt < 13; ++nt) {
    int ncol = nt * 16 + l15;
    if (ncol > NK - 1) ncol = NK - 1;
    const float* krow = kp + (size_t)ncol * E_;
#pragma unroll
    for (int kk = 0; kk < 16; ++kk) {
      v2f a, bf;
      a.x  = aq[2 * kk];
      a.y  = aq[2 * kk + 1];
      bf.x = krow[4 * kk + ko] + tek[4 * kk + ko];
      bf.y = krow[4 * kk + ko + 1] + tek[4 * kk + ko + 1];
      acc[nt] = wmma4(a, bf, acc[nt]);
    }
  }
  // + patch_mask (broadcast over q,h), mask pad columns (>=196)
#pragma unroll
  for (int nt = 0; nt < 13; ++nt) {
    int ncol = nt * 16 + l15;
    bool valid = ncol < NK;
    float pmv = valid ? pm[ncol] : 0.f;
#pragma unroll
    for (int i = 0; i < 8; ++i)
      acc[nt][i] = valid ? (acc[nt][i] + pmv) : NEG_INF;
  }

  // ---- softmax over 196 keys ----
#pragma unroll
  for (int i = 0; i < 8; ++i) {
    float mx = acc[0][i];
#pragma unroll
    for (int nt = 1; nt < 13; ++nt) mx = fmaxf(mx, acc[nt][i]);
#pragma unroll
    for (int off = 8; off > 0; off >>= 1) mx = fmaxf(mx, __shfl_xor(mx, off, 32));
    float s = 0.f;
#pragma unroll
    for (int nt = 0; nt < 13; ++nt) {
      float e = __expf(acc[nt][i] - mx);
      acc[nt][i] = e;
      s += e;
    }
#pragma unroll
    for (int off = 8; off > 0; off >>= 1) s += __shfl_xor(s, off, 32);
    float rinv = 1.f / s;
#pragma unroll
    for (int nt = 0; nt < 13; ++nt) acc[nt][i] *= rinv;
  }

  // ---- mix = P @ (v + te_v) ----
  v8f mc[4];
#pragma unroll
  for (int ct = 0; ct < 4; ++ct) mc[ct] = zero;
#pragma unroll
  for (int jt = 0; jt < 13; ++jt) {
#pragma unroll
    for (int i = 0; i < 8; ++i)
      Pl[(half * 8 + i) * 17 + l15] = acc[jt][i];
    asm volatile("s_wait_dscnt 0" ::: "memory");
#pragma unroll
    for (int ct = 0; ct < 4; ++ct) {
      const float* vcol = vp + ct * 16 + l15;
      float tv = tev[ct * 16 + l15];
#pragma unroll
      for (int kk = 0; kk < 4; ++kk) {
        int j0  = jt * 16 + kk * 4 + ko;
        int j0c = (j0 > NK - 1) ? (NK - 1) : j0;
        int j1c = (j0 + 1 > NK - 1) ? (NK - 1) : (j0 + 1);
        v2f a, bf;
        a.x  = Pl[l15 * 17 + kk * 4 + ko];
        a.y  = Pl[l15 * 17 + kk * 4 + ko + 1];
        bf.x = vcol[(size_t)j0c * E_] + tv;
        bf.y = vcol[(size_t)j1c * E_] + tv;
        mc[ct] = wmma4(a, bf, mc[ct]);
      }
    }
    asm volatile("s_wait_dscnt 0" ::: "memory");
  }

  // store: smix[(b*QS+q)*E + h*64 + c][t]
#pragma unroll
  for (int ct = 0; ct < 4; ++ct) {
#pragma unroll
    for (int i = 0; i < 8; ++i) {
      int qrow = half * 8 + i;
      smix[((size_t)(b * QS_ + qrow) * E_ + h * 64 + ct * 16 + l15) * T_ + t] =
          mc[ct][i];
    }
  }
}

// =====================================================================
// Depthwise conv over T (k=3, pad=1) + bias + residual, then mean over T.
// smix: (B*QS, E, T) row-major -> red: (B*QS, E)
// =====================================================================
__global__ void conv_mean_kernel(const float* __restrict__ smix,
                                 const float* __restrict__ cw,
                                 const float* __restrict__ cb,
                                 float* __restrict__ red) {
  int idx = blockIdx.x * blockDim.x + threadIdx.x;
  if (idx >= 128 * 768) return;
  int e = idx % 768;
  const float* xp = smix + (size_t)idx * T_;
  float x[T_];
#pragma unroll
  for (int t = 0; t < T_; ++t) x[t] = xp[t];
  float w0 = cw[e * 3 + 0], w1 = cw[e * 3 + 1], w2 = cw[e * 3 + 2];
  float bb = cb[e];
  float s = 0.f;
#pragma unroll
  for (int t = 0; t < T_; ++t) {
    float xm = (t > 0) ? x[t - 1] : 0.f;
    float xq = (t < T_ - 1) ? x[t + 1] : 0.f;
    s += w0 * xm + w1 * x[t] + w2 * xq + bb + x[t];
  }
  red[idx] = s * (1.f / T_);
}

// =====================================================================
extern "C" void kernel_launch(void* const* d_in, const int* in_sizes, int n_in,
                              void* d_out, int out_size, void* d_ws,
                              size_t ws_size, hipStream_t stream) {
  const float* x      = (const float*)d_in[0];
  const float* s      = (const float*)d_in[1];
  const float* te     = (const float*)d_in[2];
  const float* pmask  = (const float*)d_in[3];
  const float* w_in   = (const float*)d_in[4];
  const float* b_in   = (const float*)d_in[5];
  const float* out_w  = (const float*)d_in[6];
  const float* out_b  = (const float*)d_in[7];
  const float* conv_w = (const float*)d_in[8];
  const float* conv_b = (const float*)d_in[9];

  float* out = (float*)d_out;
  const size_t SZ  = (size_t)MX_ * E_;          // 9,682,944
  const size_t SSZ = (size_t)B_ * QS_ * E_;     // 98,304
  float* oq    = out;
  float* okk   = out + SZ;
  float* ov    = out + 2 * SZ;
  float* oo    = out + 3 * SZ;
  float* osq   = out + 4 * SZ;
  float* osk   = osq + SSZ;
  float* osv   = osk + SSZ;
  float* osout = osv + SSZ;

  float* ws    = (float*)d_ws;
  float* te_zz = ws;                                   // 8*2304
  float* mixw  = te_zz + (size_t)8 * 2304;             // 12608*768
  float* smixw = mixw + SZ;                            // 128*768*8
  float* sredw = smixw + (size_t)128 * 768 * T_;       // 128*768

  dim3 blk(256);
  // 1) qkv(x) -> q/k/v regions of d_out
  gemm_bias_wmma<<<dim3(36, 99), blk, 0, stream>>>(
      x, w_in, b_in, MX_, 3 * E_, E_, oq, okk, ov, E_, E_);
  // 2) qkv(s) -> s_q/s_k/s_v regions
  gemm_bias_wmma<<<dim3(36, 1), blk, 0, stream>>>(
      s, w_in, b_in, B_ * QS_, 3 * E_, E_, osq, osk, osv, E_, E_);
  // 3) qkv(te) -> workspace (single region, ld = 2304)
  gemm_bias_wmma<<<dim3(36, 1), blk, 0, stream>>>(
      te, w_in, b_in, T_, 3 * E_, E_, te_zz, nullptr, nullptr, 3 * E_, 3 * E_);
  // 4) self-attention -> mix
  attn_kernel<<<dim3(B_ * T_ * H_), blk, 0, stream>>>(oq, okk, ov, mixw);
  // 5) s-attention -> smix (conv layout)
  s_attn_kernel<<<dim3(B_ * T_ * H_), dim3(32), 0, stream>>>(
      osq, okk, ov, te_zz, pmask, smixw);
  // 6) depthwise conv + residual + mean over T
  conv_mean_kernel<<<dim3((128 * 768 + 255) / 256), blk, 0, stream>>>(
      smixw, conv_w, conv_b, sredw);
  // 7) out projection: mix @ out_w^T + out_b
  gemm_bias_wmma<<<dim3(12, 99), blk, 0, stream>>>(
      mixw, out_w, out_b, MX_, E_, E_, oo, nullptr, nullptr, E_, E_);
  // 8) s_out projection
  gemm_bias_wmma<<<dim3(12, 1), blk, 0, stream>>>(
      sredw, out_w, out_b, B_ * QS_, E_, E_, osout, nullptr, nullptr, E_, E_);
}